// MambaBlock_6090263625997
// MI455X (gfx1250) — compile-verified
//
#include <hip/hip_runtime.h>
#include <hip/hip_bf16.h>
#include <math.h>

typedef float v2f __attribute__((ext_vector_type(2)));
typedef float v8f __attribute__((ext_vector_type(8)));

#define D_MODEL 1024
#define D_STATE 16
#define D_INNER 2048
#define KCONV   4
#define BB      2
#define LL      2048
#define NTOK    (BB * LL) /* 4096 */

__device__ __forceinline__ float silu_f(float x) { return x / (1.0f + __expf(-x)); }

__device__ __forceinline__ float epilogue_f(float s, float b, int e) {
    s += b;
    if (e == 1) { // softplus
        s = (s > 20.0f) ? s : log1pf(__expf(s));
    }
    return s;
}

// ---------------------------------------------------------------------------
// C[M,N] = A[M,K] * B[K,N] (+bias) (+optional softplus), fp32 via
// V_WMMA_F32_16X16X4_F32. One wave computes a 32x32 tile (4 accumulators),
// so each fragment pair feeds 4 WMMAs. M is always a multiple of 32 here;
// N may be 16 (W_B / W_C) so column accesses are clamped/guarded.
// ---------------------------------------------------------------------------
__global__ __launch_bounds__(256) void gemm_f32_wmma(
    const float* __restrict__ A, const float* __restrict__ Bw,
    const float* __restrict__ bias, float* __restrict__ C,
    int M, int N, int K, int epi)
{
    const int lane = threadIdx.x & 31;
    const int wave = blockIdx.x * (blockDim.x >> 5) + (threadIdx.x >> 5);
    const int tiles_n = (N + 31) >> 5;
    const int tiles_m = (M + 31) >> 5;
    if (wave >= tiles_m * tiles_n) return;           // wave-uniform: EXEC stays all-1s
    const int tm = (wave / tiles_n) << 5;
    const int tn = (wave % tiles_n) << 5;

    const int half = lane >> 4;   // 0: K=0,1 slice  1: K=2,3 slice
    const int idx  = lane & 15;   // row (A) / col (B,C)

    const float* a0p = A + (size_t)(tm + idx)      * K;
    const float* a1p = A + (size_t)(tm + 16 + idx) * K;

    int nc0 = tn + idx;       if (nc0 > N - 1) nc0 = N - 1;
    int nc1 = tn + 16 + idx;  if (nc1 > N - 1) nc1 = N - 1;

    v8f acc00 = {}; v8f acc01 = {}; v8f acc10 = {}; v8f acc11 = {};

#pragma unroll 2
    for (int k = 0; k < K; k += 4) {
        const int kk = k + (half << 1);
        v2f a0 = *(const v2f*)(a0p + kk);
        v2f a1 = *(const v2f*)(a1p + kk);
        v2f b0, b1;
        b0.x = Bw[(size_t)kk       * N + nc0];
        b0.y = Bw[(size_t)(kk + 1) * N + nc0];
        b1.x = Bw[(size_t)kk       * N + nc1];
        b1.y = Bw[(size_t)(kk + 1) * N + nc1];

        acc00 = __builtin_amdgcn_wmma_f32_16x16x4_f32(false, a0, false, b0, (short)0, acc00, false, false);
        acc01 = __builtin_amdgcn_wmma_f32_16x16x4_f32(false, a0, false, b1, (short)0, acc01, false, false);
        acc10 = __builtin_amdgcn_wmma_f32_16x16x4_f32(false, a1, false, b0, (short)0, acc10, false, false);
        acc11 = __builtin_amdgcn_wmma_f32_16x16x4_f32(false, a1, false, b1, (short)0, acc11, false, false);
    }

    const bool c0ok = (tn + idx)      < N;
    const bool c1ok = (tn + 16 + idx) < N;
    const float bv0 = (bias && c0ok) ? bias[tn + idx]      : 0.0f;
    const float bv1 = (bias && c1ok) ? bias[tn + 16 + idx] : 0.0f;

#pragma unroll
    for (int v = 0; v < 8; ++v) {
        const int r0 = tm + (half << 3) + v;   // C/D layout: vgpr v -> M = v + 8*half
        const int r1 = r0 + 16;
        if (c0ok) {
            C[(size_t)r0 * N + tn + idx] = epilogue_f(acc00[v], bv0, epi);
            C[(size_t)r1 * N + tn + idx] = epilogue_f(acc10[v], bv0, epi);
        }
        if (c1ok) {
            C[(size_t)r0 * N + tn + 16 + idx] = epilogue_f(acc01[v], bv1, epi);
            C[(size_t)r1 * N + tn + 16 + idx] = epilogue_f(acc11[v], bv1, epi);
        }
    }
}

// ---------------------------------------------------------------------------
// Causal depthwise conv (K=4) + SiLU. Coalesced across d (innermost dim).
// ---------------------------------------------------------------------------
__global__ __launch_bounds__(256) void conv_silu(
    const float* __restrict__ xz, const float* __restrict__ cw,
    const float* __restrict__ cb, float* __restrict__ act, int total)
{
    const int i = blockIdx.x * blockDim.x + threadIdx.x;
    if (i >= total) return;
    const int d = i % D_INNER;
    const int l = (i / D_INNER) % LL;
    float s = cb[d];
#pragma unroll
    for (int k = 0; k < KCONV; ++k) {
        const int lt = l - (KCONV - 1) + k;
        if (lt >= 0) s += cw[d * KCONV + k] * xz[(size_t)i + (size_t)((long)(lt - l) * D_INNER)];
    }
    act[i] = silu_f(s);
}

// ---------------------------------------------------------------------------
// Selective scan. One half-wave (16 lanes) owns one (b,d) channel; lane n
// holds state n. Serial over L with a 2-stage software pipeline so the next
// step's (L2-resident) loads overlap exp/fma. y = sum_n h*C + x*D_skip done
// via a 4-level shfl_xor tree confined to the half-wave.
// ---------------------------------------------------------------------------
__global__ __launch_bounds__(256) void ssm_scan(
    const float* __restrict__ delta, const float* __restrict__ xact,
    const float* __restrict__ Bm, const float* __restrict__ Cm,
    const float* __restrict__ A_log, const float* __restrict__ D_skip,
    float* __restrict__ y)
{
    const int wave = blockIdx.x * (blockDim.x >> 5) + (threadIdx.x >> 5);
    const int lane = threadIdx.x & 31;
    const int half = lane >> 4;
    const int n    = lane & 15;
    const int total = BB * (D_INNER / 2);
    if (wave >= total) return;
    const int b = wave / (D_INNER / 2);
    const int d = (wave % (D_INNER / 2)) * 2 + half;

    const float Acoef = -__expf(A_log[d * D_STATE + n]);
    const float dsk   = D_skip[d];

    const float* dp = delta + (size_t)b * LL * D_INNER + d;
    const float* xp = xact  + (size_t)b * LL * D_INNER + d;
    const float* bp = Bm    + (size_t)b * LL * D_STATE + n;
    const float* cp = Cm    + (size_t)b * LL * D_STATE + n;
    float*       yp = y     + (size_t)b * LL * D_INNER + d;

    float h = 0.0f;
    float dt = dp[0], xt = xp[0], bt = bp[0], ct = cp[0];

    for (int t = 0; t < LL; ++t) {
        float dt2 = 0.f, xt2 = 0.f, bt2 = 0.f, ct2 = 0.f;
        if (t + 1 < LL) {
            const size_t od = (size_t)(t + 1) * D_INNER;
            const size_t ob = (size_t)(t + 1) * D_STATE;
            dt2 = dp[od]; xt2 = xp[od]; bt2 = bp[ob]; ct2 = cp[ob];
        }
        h = __expf(dt * Acoef) * h + (dt * xt) * bt;
        float v = h * ct;
        v += __shfl_xor(v, 1, 32);
        v += __shfl_xor(v, 2, 32);
        v += __shfl_xor(v, 4, 32);
        v += __shfl_xor(v, 8, 32);
        if (n == 0) yp[(size_t)t * D_INNER] = v + xt * dsk;
        dt = dt2; xt = xt2; bt = bt2; ct = ct2;
    }
}

// ---------------------------------------------------------------------------
// y *= silu(gate), in place.
// ---------------------------------------------------------------------------
__global__ __launch_bounds__(256) void gate_mul(
    float* __restrict__ y, const float* __restrict__ g, int total)
{
    const int i = blockIdx.x * blockDim.x + threadIdx.x;
    if (i < total) y[i] *= silu_f(g[i]);
}

// ---------------------------------------------------------------------------
extern "C" void kernel_launch(void* const* d_in, const int* in_sizes, int n_in,
                              void* d_out, int out_size, void* d_ws, size_t ws_size,
                              hipStream_t stream) {
    const float* x      = (const float*)d_in[0];
    const float* W_in   = (const float*)d_in[1];
    const float* b_in   = (const float*)d_in[2];
    const float* W_gate = (const float*)d_in[3];
    const float* b_gate = (const float*)d_in[4];
    const float* conv_w = (const float*)d_in[5];
    const float* conv_b = (const float*)d_in[6];
    const float* A_log  = (const float*)d_in[7];
    const float* D_skip = (const float*)d_in[8];
    const float* W_delta= (const float*)d_in[9];
    const float* b_delta= (const float*)d_in[10];
    const float* W_B    = (const float*)d_in[11];
    const float* W_C    = (const float*)d_in[12];
    const float* W_out  = (const float*)d_in[13];
    const float* b_out  = (const float*)d_in[14];
    float* out = (float*)d_out;

    const size_t big = (size_t)NTOK * D_INNER;          // 8.39M floats
    float* ws   = (float*)d_ws;
    float* xz   = ws;                                   // later reused as delta
    float* xg   = xz  + big;
    float* act  = xg  + big;
    float* yb   = act + big;
    float* Bmat = yb  + big;
    float* Cmat = Bmat + (size_t)NTOK * D_STATE;

    auto launch_gemm = [&](const float* A, const float* Bw, const float* bias,
                           float* C, int M, int N, int K, int e) {
        const int tiles = ((M + 31) / 32) * ((N + 31) / 32);
        const int waves_per_block = 8;                  // 256 threads
        const int nb = (tiles + waves_per_block - 1) / waves_per_block;
        gemm_f32_wmma<<<nb, 256, 0, stream>>>(A, Bw, bias, C, M, N, K, e);
    };

    // 1-2: input + gate projections
    launch_gemm(x, W_in,   b_in,   xz, NTOK, D_INNER, D_MODEL, 0);
    launch_gemm(x, W_gate, b_gate, xg, NTOK, D_INNER, D_MODEL, 0);

    // 3: causal depthwise conv + SiLU
    {
        const int total = NTOK * D_INNER;
        conv_silu<<<(total + 255) / 256, 256, 0, stream>>>(xz, conv_w, conv_b, act, total);
    }

    // 4: delta = softplus(act @ W_delta + b_delta)   (reuses xz buffer)
    launch_gemm(act, W_delta, b_delta, xz, NTOK, D_INNER, D_INNER, 1);

    // 5-6: B and C projections (N = 16)
    launch_gemm(act, W_B, nullptr, Bmat, NTOK, D_STATE, D_INNER, 0);
    launch_gemm(act, W_C, nullptr, Cmat, NTOK, D_STATE, D_INNER, 0);

    // 7: selective scan (+ D_skip fused)
    {
        const int swaves  = BB * (D_INNER / 2);         // 2048 waves
        const int threads = swaves * 32;
        ssm_scan<<<(threads + 255) / 256, 256, 0, stream>>>(xz, act, Bmat, Cmat,
                                                            A_log, D_skip, yb);
    }

    // 8: gating
    {
        const int total = NTOK * D_INNER;
        gate_mul<<<(total + 255) / 256, 256, 0, stream>>>(yb, xg, total);
    }

    // 9: output projection
    launch_gemm(yb, W_out, b_out, out, NTOK, D_MODEL, D_INNER, 0);
}